// EigenEmbedding_55508157333879
// MI455X (gfx1250) — compile-verified
//
#include <hip/hip_runtime.h>

// Batched symmetric eigendecomposition (eigh) via two-sided block Jacobi.
// B=256 matrices, N=512. Block size 16 -> 32 block columns, 16 disjoint
// pairs per round (round-robin ordering), 31 rounds per sweep, 6 sweeps.
// All O(N^3) updates go through V_WMMA_F32_16X16X4_F32 (fp32 tensor path).
// GEMM kernels keep J-side fragments in registers across 8 chunks of the
// long dimension (64 WMMAs/wave), double-buffer chunk staging in LDS, and
// software-pipeline the next chunk's b128 loads through registers so the
// load wait lands after the WMMA burst.

#define NMAT 256
#define NDIM 512
#define NBLK 32          // number of 16-wide block columns
#define PAIRS 16
#define ROUNDS 31
#define SWEEPS 6
#define INNER_SWEEPS 4   // inner Jacobi sweeps on the 32x32 subproblem
#define JPAD 36          // LDS row pad: 36 floats = 144B (16B multiple)

typedef __attribute__((ext_vector_type(2))) float v2f;
typedef __attribute__((ext_vector_type(4))) float v4f;
typedef __attribute__((ext_vector_type(8))) float v8f;

// Round-robin (circle method) disjoint pairing of 32 items, 31 rounds.
__device__ __forceinline__ void pair_of(int r, int k, int& p, int& q) {
    const int m = NBLK - 1; // 31
    int a, b;
    if (k == 0) { a = NBLK - 1; b = r % m; }
    else        { a = (r + k) % m; b = (r + m - k) % m; }
    p = a < b ? a : b;
    q = a < b ? b : a;
}

// Map local index 0..31 to global row/col for block pair (p,q).
__device__ __forceinline__ int bmap(int i, int p, int q) {
    return (i < 16) ? (p * 16 + i) : (q * 16 + (i - 16));
}

// ---------------------------------------------------------------- init ----
__global__ void k_init(const float* __restrict__ A, float* __restrict__ Aw,
                       float* __restrict__ V) {
    size_t i4 = (size_t)blockIdx.x * blockDim.x + threadIdx.x;
    const v4f* A4 = (const v4f*)A;
    v4f* Aw4 = (v4f*)Aw;
    v4f* V4 = (v4f*)V;
    Aw4[i4] = A4[i4];
    int rc = (int)((i4 * 4) % (size_t)(NDIM * NDIM));
    int row = rc >> 9, col = rc & 511; // 4 consecutive cols, same row
    v4f id;
    id.x = (col + 0 == row) ? 1.0f : 0.0f;
    id.y = (col + 1 == row) ? 1.0f : 0.0f;
    id.z = (col + 2 == row) ? 1.0f : 0.0f;
    id.w = (col + 3 == row) ? 1.0f : 0.0f;
    V4[i4] = id;
}

// ------------------------------------------------- 32x32 subproblem -------
// One workgroup (256 thr) per (matrix, pair): parallel element-Jacobi on the
// 32x32 pivot submatrix in LDS; writes the accumulated rotation J (32x32).
__global__ void k_solve(const float* __restrict__ Aw, float* __restrict__ Jbuf,
                        int round) {
    __shared__ float S[32][33];
    __shared__ float Jm[32][33];
    __shared__ float cbuf[16], sbuf[16];

    int mat = blockIdx.x, pr = blockIdx.y;
    int p, q;
    pair_of(round, pr, p, q);
    const float* Ab = Aw + (size_t)mat * NDIM * NDIM;
    int tid = threadIdx.x;

    for (int idx = tid; idx < 32 * 32; idx += 256) {
        int r = idx >> 5, c = idx & 31;
        S[r][c]  = Ab[(size_t)bmap(r, p, q) * NDIM + bmap(c, p, q)];
        Jm[r][c] = (r == c) ? 1.0f : 0.0f;
    }
    __syncthreads();

    for (int sw = 0; sw < INNER_SWEEPS; ++sw) {
        for (int r = 0; r < 31; ++r) {
            if (tid < 16) {
                int i, j;
                pair_of(r, tid, i, j);
                float app = S[i][i], aqq = S[j][j], apq = S[i][j];
                float cc = 1.0f, ss = 0.0f;
                if (__builtin_fabsf(apq) > 1e-30f) {
                    float tau = (aqq - app) / (2.0f * apq);
                    float rt  = __builtin_sqrtf(1.0f + tau * tau);
                    float t   = (tau >= 0.0f) ? 1.0f / (tau + rt)
                                              : 1.0f / (tau - rt);
                    cc = 1.0f / __builtin_sqrtf(1.0f + t * t);
                    ss = t * cc;
                }
                cbuf[tid] = cc;
                sbuf[tid] = ss;
            }
            __syncthreads();
            // Row phase: S <- G^T S  (disjoint rows across the 16 pairs)
            for (int it = 0; it < 2; ++it) {
                int idx = tid + it * 256;
                int k = idx >> 5, t = idx & 31;
                int i, j;
                pair_of(r, k, i, j);
                float cc = cbuf[k], ss = sbuf[k];
                float x = S[i][t], y = S[j][t];
                S[i][t] = cc * x - ss * y;
                S[j][t] = ss * x + cc * y;
            }
            __syncthreads();
            // Column phase: S <- S G, J <- J G (disjoint columns)
            for (int it = 0; it < 2; ++it) {
                int idx = tid + it * 256;
                int k = idx >> 5, t = idx & 31;
                int i, j;
                pair_of(r, k, i, j);
                float cc = cbuf[k], ss = sbuf[k];
                float x = S[t][i], y = S[t][j];
                S[t][i] = cc * x - ss * y;
                S[t][j] = ss * x + cc * y;
                float jx = Jm[t][i], jy = Jm[t][j];
                Jm[t][i] = cc * jx - ss * jy;
                Jm[t][j] = ss * jx + cc * jy;
            }
            __syncthreads();
        }
    }

    float* Jd = Jbuf + ((size_t)mat * PAIRS + pr) * 1024;
    for (int idx = tid; idx < 1024; idx += 256)
        Jd[idx] = Jm[idx >> 5][idx & 31];
}

// -------------------------------------------- row update: A <- J^T A ------
// One workgroup per (pair, matrix), 256 thr = 8 waves: mtile(2) x ntile(4).
// a-frags (J^T) live in registers; 8 column chunks of 64, double-buffered
// transposed staging, next chunk pipelined through registers.
__global__ void k_rowupd(float* __restrict__ Aw, const float* __restrict__ Jbuf,
                         int round) {
    __shared__ __align__(16) float Jt[32][JPAD];    // Jt[m][k] = J[k][m]
    __shared__ __align__(16) float Tt[2][64][JPAD]; // Tt[b][n][k] = A[row(k)][cb+n]

    int pr = blockIdx.x, mat = blockIdx.y;
    int p, q;
    pair_of(round, pr, p, q);
    const float* Jd = Jbuf + ((size_t)mat * PAIRS + pr) * 1024;
    float* Ab = Aw + (size_t)mat * NDIM * NDIM;
    int tid = threadIdx.x;
    int lane = tid & 31, wave = tid >> 5;
    int mtile = wave & 1, ntile = wave >> 1;
    int h = lane >> 4, l = lane & 15;
    int m = mtile * 16 + l;

    // Staging mapping: each thread owns (krow, n4) and covers k=krow (p rows)
    // and k=krow+16 (q rows) with one b128 load each. Fully coalesced.
    int krow = tid >> 4;          // 0..15
    int n4 = (tid & 15) * 4;      // 0..60
    const float* s0 = Ab + (size_t)(p * 16 + krow) * NDIM + n4;
    const float* s1 = Ab + (size_t)(q * 16 + krow) * NDIM + n4;

    // Stage J transposed (coalesced global read).
    for (int idx = tid; idx < 1024; idx += 256)
        Jt[idx & 31][idx >> 5] = Jd[idx];

    // Preload chunk 0 into buffer 0.
    {
        v4f x0 = *(const v4f*)(s0);
        v4f x1 = *(const v4f*)(s1);
        Tt[0][n4 + 0][krow] = x0.x; Tt[0][n4 + 1][krow] = x0.y;
        Tt[0][n4 + 2][krow] = x0.z; Tt[0][n4 + 3][krow] = x0.w;
        Tt[0][n4 + 0][krow + 16] = x1.x; Tt[0][n4 + 1][krow + 16] = x1.y;
        Tt[0][n4 + 2][krow + 16] = x1.z; Tt[0][n4 + 3][krow + 16] = x1.w;
    }
    __syncthreads();

    // Load the 8 a-fragments once; constant across all column chunks.
    v2f aa[8];
#pragma unroll
    for (int kk = 0; kk < 8; ++kk)
        aa[kk] = *reinterpret_cast<const v2f*>(&Jt[m][kk * 4 + 2 * h]);

    // Output row base for this wave (mtile uniform -> scalar select).
    float* drow = Ab + (size_t)((mtile ? q : p) * 16 + 8 * h) * NDIM
                     + ntile * 16 + l;

    for (int ct = 0; ct < 8; ++ct) {
        int cur = ct & 1;
        v4f y0, y1;
        if (ct < 7) { // issue next chunk's loads before the WMMA burst
            y0 = *(const v4f*)(s0 + (ct + 1) * 64);
            y1 = *(const v4f*)(s1 + (ct + 1) * 64);
            if (ct < 6) {
                __builtin_prefetch(s0 + (ct + 2) * 64, 0, 1);
                __builtin_prefetch(s1 + (ct + 2) * 64, 0, 1);
            }
        }

        v8f acc = {};
#pragma unroll
        for (int kk = 0; kk < 8; ++kk) {
            v2f b = *reinterpret_cast<const v2f*>(&Tt[cur][ntile * 16 + l][kk * 4 + 2 * h]);
            acc = __builtin_amdgcn_wmma_f32_16x16x4_f32(false, aa[kk], false, b,
                                                        (short)0, acc, false, false);
        }
#pragma unroll
        for (int v = 0; v < 8; ++v)
            drow[(size_t)v * NDIM + ct * 64] = acc[v];

        if (ct < 7) { // spill pipelined registers into the other buffer
            int nxt = cur ^ 1;
            Tt[nxt][n4 + 0][krow] = y0.x; Tt[nxt][n4 + 1][krow] = y0.y;
            Tt[nxt][n4 + 2][krow] = y0.z; Tt[nxt][n4 + 3][krow] = y0.w;
            Tt[nxt][n4 + 0][krow + 16] = y1.x; Tt[nxt][n4 + 1][krow + 16] = y1.y;
            Tt[nxt][n4 + 2][krow + 16] = y1.z; Tt[nxt][n4 + 3][krow + 16] = y1.w;
        }
        __syncthreads();
    }
}

// ------------------------- column update: A <- A J  and  V <- V J ---------
// grid (2, pairs, mats): x=0 -> A, x=1 -> V. b-frags (J) in registers;
// 8 row chunks of 64, double-buffered staging with b128 LDS stores.
__global__ void k_colupd(float* __restrict__ Aw, float* __restrict__ V,
                         const float* __restrict__ Jbuf, int round) {
    __shared__ __align__(16) float Jn[32][JPAD];   // Jn[n][k] = J[k][n]
    __shared__ __align__(16) float T[2][64][JPAD]; // 64-row x 32-col stripe

    int which = blockIdx.x, pr = blockIdx.y, mat = blockIdx.z;
    int p, q;
    pair_of(round, pr, p, q);
    float* Mb = ((which == 0) ? Aw : V) + (size_t)mat * NDIM * NDIM;
    const float* Jd = Jbuf + ((size_t)mat * PAIRS + pr) * 1024;
    int tid = threadIdx.x;
    int lane = tid & 31, wave = tid >> 5;
    int mtile = wave >> 1, ntile = wave & 1; // 4 x 2 tiles per 64-row chunk
    int h = lane >> 4, l = lane & 15;

    // Staging mapping: thread owns (r, c4); one b128 from the p-block and one
    // from the q-block of its row; b128 stores into LDS (144B row stride).
    int r = tid >> 2;             // 0..63
    int c4 = (tid & 3) * 4;       // 0,4,8,12
    const float* s0 = Mb + (size_t)r * NDIM + p * 16 + c4;
    const float* s1 = Mb + (size_t)r * NDIM + q * 16 + c4;

    for (int idx = tid; idx < 1024; idx += 256)
        Jn[idx & 31][idx >> 5] = Jd[idx];

    {
        v4f x0 = *(const v4f*)(s0);
        v4f x1 = *(const v4f*)(s1);
        *(v4f*)&T[0][r][c4] = x0;
        *(v4f*)&T[0][r][16 + c4] = x1;
    }
    __syncthreads();

    v2f bb[8];
#pragma unroll
    for (int kk = 0; kk < 8; ++kk)
        bb[kk] = *reinterpret_cast<const v2f*>(&Jn[ntile * 16 + l][kk * 4 + 2 * h]);

    float* drow = Mb + (size_t)(mtile * 16 + 8 * h) * NDIM
                     + (ntile ? q : p) * 16 + l;

    for (int rt = 0; rt < 8; ++rt) {
        int cur = rt & 1;
        v4f y0, y1;
        if (rt < 7) {
            y0 = *(const v4f*)(s0 + (size_t)(rt + 1) * 64 * NDIM);
            y1 = *(const v4f*)(s1 + (size_t)(rt + 1) * 64 * NDIM);
            if (rt < 6) {
                __builtin_prefetch(s0 + (size_t)(rt + 2) * 64 * NDIM, 0, 1);
                __builtin_prefetch(s1 + (size_t)(rt + 2) * 64 * NDIM, 0, 1);
            }
        }

        v8f acc = {};
#pragma unroll
        for (int kk = 0; kk < 8; ++kk) {
            v2f a = *reinterpret_cast<const v2f*>(&T[cur][mtile * 16 + l][kk * 4 + 2 * h]);
            acc = __builtin_amdgcn_wmma_f32_16x16x4_f32(false, a, false, bb[kk],
                                                        (short)0, acc, false, false);
        }
#pragma unroll
        for (int v = 0; v < 8; ++v)
            drow[(size_t)(rt * 64 + v) * NDIM] = acc[v];

        if (rt < 7) {
            int nxt = cur ^ 1;
            *(v4f*)&T[nxt][r][c4] = y0;
            *(v4f*)&T[nxt][r][16 + c4] = y1;
        }
        __syncthreads();
    }
}

// ------------------------------------- eigenvalue ranking (ascending) -----
__global__ void k_rank(const float* __restrict__ Aw, float* __restrict__ eig,
                       int* __restrict__ perm) {
    __shared__ float d[NDIM];
    int mat = blockIdx.x, tid = threadIdx.x; // 512 threads
    float v = Aw[(size_t)mat * NDIM * NDIM + (size_t)tid * NDIM + tid];
    d[tid] = v;
    __syncthreads();
    int rank = 0;
    for (int k = 0; k < NDIM; ++k) {
        float dk = d[k];
        rank += (dk < v) || (dk == v && k < tid);
    }
    perm[mat * NDIM + rank] = tid; // source column for ascending position
    eig[mat * NDIM + tid] = v;
}

// --------------------------- permute V columns into workspace buffer ------
__global__ void k_perm(const float* __restrict__ V, float* __restrict__ Vt,
                       const int* __restrict__ perm) {
    int row = blockIdx.x, mat = blockIdx.y, j = threadIdx.x;
    size_t base = ((size_t)mat * NDIM + row) * NDIM;
    Vt[base + j] = V[base + perm[mat * NDIM + j]];
}

__global__ void k_copy(const float* __restrict__ src, float* __restrict__ dst) {
    size_t i4 = (size_t)blockIdx.x * blockDim.x + threadIdx.x;
    ((v4f*)dst)[i4] = ((const v4f*)src)[i4];
}

// --------------------------------------------------------------------------
extern "C" void kernel_launch(void* const* d_in, const int* in_sizes, int n_in,
                              void* d_out, int out_size, void* d_ws, size_t ws_size,
                              hipStream_t stream) {
    const float* A = (const float*)d_in[0];
    float* V = (float*)d_out;

    // Workspace layout: Awork | Jbuf | eig | perm
    float* Aw = (float*)d_ws;
    size_t awElems = (size_t)NMAT * NDIM * NDIM;           // 256 MB
    float* Jbuf = Aw + awElems;                            // 16 MB
    size_t jElems = (size_t)NMAT * PAIRS * 32 * 32;
    float* eig = Jbuf + jElems;                            // 512 KB
    int* perm = (int*)(eig + (size_t)NMAT * NDIM);         // 512 KB

    dim3 blk(256);
    int nblk_v4 = (int)(awElems / 4 / 256);

    k_init<<<dim3(nblk_v4), blk, 0, stream>>>(A, Aw, V);

    for (int s = 0; s < SWEEPS; ++s) {
        for (int r = 0; r < ROUNDS; ++r) {
            k_solve <<<dim3(NMAT, PAIRS),    blk, 0, stream>>>(Aw, Jbuf, r);
            k_rowupd<<<dim3(PAIRS, NMAT),    blk, 0, stream>>>(Aw, Jbuf, r);
            k_colupd<<<dim3(2, PAIRS, NMAT), blk, 0, stream>>>(Aw, V, Jbuf, r);
        }
    }

    k_rank<<<dim3(NMAT), dim3(NDIM), 0, stream>>>(Aw, eig, perm);
    k_perm<<<dim3(NDIM, NMAT), dim3(NDIM), 0, stream>>>(V, Aw, perm); // Aw as temp
    k_copy<<<dim3(nblk_v4), blk, 0, stream>>>(Aw, V);
}